// aggregate_nodes_temporal_feature_5927054868791
// MI455X (gfx1250) — compile-verified
//
#include <hip/hip_runtime.h>

typedef float v2f __attribute__((ext_vector_type(2)));
typedef float v4f __attribute__((ext_vector_type(4)));
typedef float v8f __attribute__((ext_vector_type(8)));

#define T_MAX 32
#define F_DIM 128
#define NODES_PER_BLOCK 8   // 8 waves of 32 = 256 threads

// Tiny deterministic prefix-sum of num_nodes into workspace (B+1 ints).
__global__ void node_offsets_kernel(const int* __restrict__ num_nodes, int B,
                                    int* __restrict__ offs) {
    if (threadIdx.x == 0 && blockIdx.x == 0) {
        int acc = 0;
        for (int b = 0; b < B; ++b) { offs[b] = acc; acc += num_nodes[b]; }
        offs[B] = acc;
    }
}

__global__ __launch_bounds__(256)
void aggregate_temporal_kernel(const float* __restrict__ X,
                               const int* __restrict__ lengths,
                               const int* __restrict__ offs,
                               const float* __restrict__ wq,
                               float* __restrict__ out,
                               int N, int B) {
    const int lane = threadIdx.x & 31;
    const int wave = threadIdx.x >> 5;
    const int node = blockIdx.x * NODES_PER_BLOCK + wave;
    if (node >= N) return;   // wave-uniform: EXEC stays all-ones below

    // --- graph id via wave-uniform binary search over node offsets ---
    int lo = 0, hi = B - 1;
    while (lo < hi) {
        int mid = (lo + hi + 1) >> 1;
        if (offs[mid] <= node) lo = mid; else hi = mid - 1;
    }
    const int len = lengths[lo];

    const float* tile = X + (size_t)node * (T_MAX * F_DIM);

    // ================= Pass 1: scores (the HBM-facing, coalesced pass) ======
    // Lane covers f-slice [4*lane, 4*lane+4). Each iteration: one 512B
    // fully-coalesced global_load_b128 across the wave.
    const v4f q = *(const v4f*)(wq + lane * 4);
    float wv = 0.0f;   // after loop: lane t holds masked score w[t]
    #pragma unroll
    for (int t = 0; t < T_MAX; ++t) {
        v4f x = *(const v4f*)(tile + t * F_DIM + lane * 4);
        float p = x[0] * q[0] + x[1] * q[1] + x[2] * q[2] + x[3] * q[3];
        // full-wave butterfly sum -> every lane holds score[t]
        #pragma unroll
        for (int d = 16; d >= 1; d >>= 1) p += __shfl_xor(p, d, 32);
        if (lane == t && t < len) wv = p;   // select + temporal mask
    }

    // ================= Pass 2: out = w^T * X via V_WMMA_F32_16X16X4_F32 =====
    // D(16x16) = sum_k A(16x4) * B(4x16), accumulated over 8 k-steps (K=32).
    // A[m,k] = w[k0+k]  (broadcast over M)   -> rows of D all identical.
    // B[k,n] = X[k0+k, fb*16+n]              -> re-read from L0/L2 (hot).
    //
    // f32 A 16x4 layout (ISA table): lanes 0-15: v0=K0, v1=K1;
    //                                lanes 16-31: v0=K2, v1=K3.
    // f32 B 4x16 layout (row-striped): v0: lanes0-15=(K=0,N=lane),
    //   lanes16-31=(K=2,N=lane-16); v1: K=1 / K=3.
    const int hi16 = (lane >= 16) ? 2 : 0;
    const int col  = lane & 15;
    const size_t obase = (size_t)node * F_DIM;

    // A fragments are invariant across f-blocks: build once, cache in VGPRs.
    v2f Am[T_MAX / 4];
    #pragma unroll
    for (int ks = 0; ks < T_MAX / 4; ++ks) {
        const int tt = ks * 4 + hi16;
        float a0 = __shfl(wv, tt, 32);          // w[k0 + (lane<16?0:2)]
        float a1 = __shfl(wv, tt + 1, 32);      // w[k0 + (lane<16?1:3)]
        Am[ks][0] = a0;
        Am[ks][1] = a1;
    }

    #pragma unroll
    for (int fb = 0; fb < F_DIM / 16; ++fb) {
        v8f acc = {};
        #pragma unroll
        for (int ks = 0; ks < T_MAX / 4; ++ks) {
            const int tt = ks * 4 + hi16;
            float b0 = tile[tt * F_DIM + fb * 16 + col];
            float b1 = tile[(tt + 1) * F_DIM + fb * 16 + col];
            v2f Bm = {b0, b1};
            acc = __builtin_amdgcn_wmma_f32_16x16x4_f32(
                false, Am[ks], false, Bm, (short)0, acc, false, false);
        }
        // All D rows equal -> acc[0] holds out[fb*16 + lane%16] in every lane.
        if (lane < 16) {
            __builtin_nontemporal_store(acc[0], &out[obase + fb * 16 + lane]);
        }
    }
}

extern "C" void kernel_launch(void* const* d_in, const int* in_sizes, int n_in,
                              void* d_out, int out_size, void* d_ws, size_t ws_size,
                              hipStream_t stream) {
    const float* X         = (const float*)d_in[0];   // (N, 32, 128) f32
    const int*   lengths   = (const int*)  d_in[1];   // (B,)
    const int*   num_nodes = (const int*)  d_in[2];   // (B,)
    const float* wq        = (const float*)d_in[3];   // (128,)
    float*       out       = (float*)d_out;           // (N, 128) f32

    const int B = in_sizes[1];
    const int N = out_size / F_DIM;

    int* offs = (int*)d_ws;   // B+1 ints of scratch
    node_offsets_kernel<<<1, 32, 0, stream>>>(num_nodes, B, offs);

    const int blocks = (N + NODES_PER_BLOCK - 1) / NODES_PER_BLOCK;
    aggregate_temporal_kernel<<<blocks, 256, 0, stream>>>(
        X, lengths, offs, wq, out, N, B);
}